// PcheckArcFaceInnerProduct_67765993997129
// MI455X (gfx1250) — compile-verified
//
#include <hip/hip_runtime.h>
#include <math.h>

// Problem constants (match reference)
#define BB 512       // batch rows (M)
#define DD 512       // feature dim (K)
#define CC 100000    // classes (N)

#define SCALE_F   30.0f
#define MARGIN_F  0.5f
#define THRESH_F  (-0.8775825618903728f)   // -cos(0.5)
#define COSM_F    (0.8775825618903728f)    // cos(0.5)
#define SINM_F    (0.4794255386042030f)    // sin(0.5)
#define MEXT_F    (-0.2397127693021015f)   // -MARGIN*sin(MARGIN)

// GEMM tiling
#define BM 512
#define BN 64
#define BK 16
#define LDA 20   // padded K-stride (floats) for A in LDS
#define LDB 20   // padded K-stride (floats) for B in LDS

typedef float v2f __attribute__((ext_vector_type(2)));
typedef float v8f __attribute__((ext_vector_type(8)));

// CDNA5 async global->LDS copies (ASYNCcnt-tracked), per CDNA5 ISA.
// Inline asm keeps this portable across ROCm-7.2 / amdgpu-toolchain clang builtins.
#define ASYNC_LDS_B128(ldsoff, gptr)                                       \
  asm volatile("global_load_async_to_lds_b128 %0, %1, off"                 \
               :: "v"(ldsoff), "v"(gptr) : "memory")
#define ASYNC_LDS_B32(ldsoff, gptr)                                        \
  asm volatile("global_load_async_to_lds_b32 %0, %1, off"                  \
               :: "v"(ldsoff), "v"(gptr) : "memory")
#define WAIT_ASYNC0() asm volatile("s_wait_asynccnt 0x0" ::: "memory")

// ---------------------------------------------------------------------------
// 1/||feat_row||  (512 rows x 512)
__global__ void feat_norm_kernel(const float* __restrict__ feat,
                                 float* __restrict__ inv_nf) {
  __shared__ float red[256];
  const int r = blockIdx.x;
  const int t = threadIdx.x;
  float a = feat[r * DD + t];
  float b = feat[r * DD + t + 256];
  red[t] = a * a + b * b;
  __syncthreads();
  for (int off = 128; off > 0; off >>= 1) {
    if (t < off) red[t] += red[t + off];
    __syncthreads();
  }
  if (t == 0) inv_nf[r] = rsqrtf(red[0]);
}

// 1/||w_row||  (100000 rows x 512); one wave32 per row, 8 rows per block
__global__ void w_norm_kernel(const float* __restrict__ w,
                              float* __restrict__ inv_nw) {
  const int tid  = blockIdx.x * blockDim.x + threadIdx.x;
  const int c    = tid >> 5;
  const int lane = tid & 31;
  if (c >= CC) return;
  const float* row = w + (size_t)c * DD;
  float s = 0.0f;
#pragma unroll
  for (int i = 0; i < 16; ++i) {
    float x = row[lane + 32 * i];
    s += x * x;
  }
#pragma unroll
  for (int off = 16; off > 0; off >>= 1) s += __shfl_xor(s, off, 32);
  if (lane == 0) inv_nw[c] = rsqrtf(s);
}

// ---------------------------------------------------------------------------
// Fused GEMM (f32 WMMA) + ArcFace epilogue.
// Block: 1024 threads = 32 waves. Tile: BM=512 x BN=64 (full-M stripe ->
// weights stream from HBM once). Double-buffered async global->LDS pipeline:
//   wait_asynccnt 0 ; barrier ; issue tile kt+1 -> buf^1 ; compute buf
// One barrier per iteration; DMA of the next tile overlaps the WMMA phase.
__launch_bounds__(1024, 1)
__global__ void arcface_fused_kernel(const float* __restrict__ feat,
                                     const float* __restrict__ w,
                                     const int*   __restrict__ label,
                                     const float* __restrict__ inv_nf,
                                     const float* __restrict__ inv_nw,
                                     float* __restrict__ out) {
  __shared__ float Asl[2][BM * LDA];   // 2 x 40 KB, [m][k] padded
  __shared__ float Bsl[2][BN * LDB];   // 2 x  5 KB, [n][k] padded
  __shared__ float s_inf[BM];
  __shared__ float s_inw[BN];
  __shared__ int   s_lab[BM];

  const int tid   = threadIdx.x;
  const int cBase = blockIdx.x * BN;

  // One-time preload of labels / inverse norms (visible after first barrier)
  if (tid < BM) {
    s_lab[tid] = label[tid];
    s_inf[tid] = inv_nf[tid];
  } else if (tid < BM + BN) {
    int n = tid - BM;
    int c = cBase + n;
    s_inw[n] = (c < CC) ? inv_nw[c] : 0.0f;
  }

  // --- async staging addresses: every thread issues 2x b128 (A) + 1x b32 (B)
  const int rowA = tid >> 2;            // 0..255
  const int kq4  = (tid & 3) * 4;       // 0,4,8,12
  const float* gA0 = feat + (size_t)rowA * DD + kq4;
  const float* gA1 = feat + (size_t)(rowA + 256) * DD + kq4;
  int cB = cBase + (tid >> 4);          // 64 rows x 16 k-elems
  if (cB >= CC) cB = CC - 1;            // clamp; stores are predicated
  const float* gB = w + (size_t)cB * DD + (tid & 15);

  // Generic pointers to __shared__ carry the LDS byte offset in their low 32b.
  const unsigned ldsA0 = (unsigned)(uintptr_t)&Asl[0][rowA * LDA + kq4];
  const unsigned ldsA1 = (unsigned)(uintptr_t)&Asl[0][(rowA + 256) * LDA + kq4];
  const unsigned ldsB  = (unsigned)(uintptr_t)&Bsl[0][(tid >> 4) * LDB + (tid & 15)];
  const unsigned AOFFB = BM * LDA * 4;  // bytes between A double-buffers
  const unsigned BOFFB = BN * LDB * 4;

  const int wave   = tid >> 5;
  const int lane   = tid & 31;
  const int laneHi = lane >> 4;         // 0: lanes 0-15, 1: lanes 16-31
  const int lane15 = lane & 15;

  const int nTile = wave & 3;           // which 16-col slab of BN
  const int mBase = (wave >> 2) * 64;   // 4 stacked 16-row tiles

  v8f acc[4];
  const v8f vzero = {0.f, 0.f, 0.f, 0.f, 0.f, 0.f, 0.f, 0.f};
#pragma unroll
  for (int t = 0; t < 4; ++t) acc[t] = vzero;

  // Prologue: tile 0 -> buffer 0
  ASYNC_LDS_B128(ldsA0, gA0);
  ASYNC_LDS_B128(ldsA1, gA1);
  ASYNC_LDS_B32(ldsB, gB);

  int buf = 0;
  for (int kt = 0; kt < DD; kt += BK) {
    WAIT_ASYNC0();        // my tile-kt copies are in LDS
    __syncthreads();      // everyone's copies in LDS; everyone done reading buf^1
    if (kt + BK < DD) {   // uniform branch: EXEC stays all-ones
      const unsigned bo = (unsigned)(buf ^ 1);
      ASYNC_LDS_B128(ldsA0 + bo * AOFFB, gA0 + kt + BK);
      ASYNC_LDS_B128(ldsA1 + bo * AOFFB, gA1 + kt + BK);
      ASYNC_LDS_B32(ldsB + bo * BOFFB, gB + kt + BK);
    }

    const float* Ab = &Asl[buf][0];
    const float* Bb = &Bsl[buf][0];
#pragma unroll
    for (int k0 = 0; k0 < BK; k0 += 4) {
      // ISA f32 fragment striping: lanes 0-15 carry K={k0,k0+1},
      // lanes 16-31 carry K={k0+2,k0+3} (VGPR0/VGPR1).
      const int kk = k0 + laneHi * 2;
      v2f bf = *(const v2f*)(Bb + (nTile * 16 + lane15) * LDB + kk);
#pragma unroll
      for (int t = 0; t < 4; ++t) {
        v2f af = *(const v2f*)(Ab + (mBase + t * 16 + lane15) * LDA + kk);
        acc[t] = __builtin_amdgcn_wmma_f32_16x16x4_f32(
            /*neg_a=*/false, af, /*neg_b=*/false, bf,
            /*c_mod=*/(short)0, acc[t], /*reuse_a=*/false, /*reuse_b=*/false);
      }
    }
    buf ^= 1;
  }

  // ---------------- fused ArcFace epilogue (closed-form, no transcendentals)
  // cos(acos(x)+M) == x*cos(M) - sin(M)*sqrt(1-x^2)  on the clamped domain.
  const size_t BC = (size_t)BB * (size_t)CC;
  float* out_cos  = out;
  float* out_marg = out + BC;
  float* out_ip   = out + 2 * BC;

  const int n = cBase + nTile * 16 + lane15;
  if (n < CC) {
    const float inw = s_inw[nTile * 16 + lane15];
#pragma unroll
    for (int t = 0; t < 4; ++t) {
#pragma unroll
      for (int v = 0; v < 8; ++v) {
        // C/D layout: VGPR v, lanes 0-15 -> M = v ; lanes 16-31 -> M = v+8
        const int m   = mBase + t * 16 + laneHi * 8 + v;
        const float ip   = acc[t][v];
        const float cosv = ip * s_inf[m] * inw;
        const float cc   = fminf(fmaxf(cosv, -1.0f), 1.0f);
        float marg = SCALE_F * cc;                 // non-label columns
        if (s_lab[m] == n) {                       // rare: ~1 element / row
          if (cosv > THRESH_F) {
            const float sn = sqrtf(fmaxf(1.0f - cc * cc, 0.0f));
            marg = SCALE_F * (cc * COSM_F - sn * SINM_F);
          } else {
            marg = SCALE_F * (cc + MEXT_F);
          }
        }
        const size_t off = (size_t)m * CC + n;
        out_cos[off]  = cosv;
        out_marg[off] = marg;
        out_ip[off]   = ip;
      }
    }
  }
}

// ---------------------------------------------------------------------------
extern "C" void kernel_launch(void* const* d_in, const int* in_sizes, int n_in,
                              void* d_out, int out_size, void* d_ws, size_t ws_size,
                              hipStream_t stream) {
  const float* feat  = (const float*)d_in[0];
  const float* wts   = (const float*)d_in[1];
  const int*   label = (const int*)d_in[2];
  float* out = (float*)d_out;

  float* inv_nf = (float*)d_ws;        // 512 floats
  float* inv_nw = inv_nf + BB;         // 100000 floats

  feat_norm_kernel<<<BB, 256, 0, stream>>>(feat, inv_nf);
  w_norm_kernel<<<(CC * 32 + 255) / 256, 256, 0, stream>>>(wts, inv_nw);

  const int nBlocks = (CC + BN - 1) / BN;   // 1563
  arcface_fused_kernel<<<nBlocks, 1024, 0, stream>>>(feat, wts, label,
                                                     inv_nf, inv_nw, out);
}